// GraphTransformer_56642028699870
// MI455X (gfx1250) — compile-verified
//
#include <hip/hip_runtime.h>
#include <math.h>

#define HCQ 128   // heads * channels
#define NHD 8     // heads
#define CCH 16    // channels per head
#define FED 7     // edge feature dim
#define RHD 500   // reghead hidden (structurally fixed by reference)

typedef __attribute__((ext_vector_type(2))) float v2f;
typedef __attribute__((ext_vector_type(8))) float v8f;

__device__ __forceinline__ float leaky(float v) { return v > 0.f ? v : 0.01f * v; }

// float atomic max via sign-split int/uint atomics (works with -inf init)
__device__ __forceinline__ void atomicMaxF32(float* addr, float v) {
  if (v >= 0.f) atomicMax((int*)addr, __float_as_int(v));
  else          atomicMin((unsigned int*)addr, (unsigned int)__float_as_int(v));
}

// ---------------- layer-1 projections: K is tiny (2), plain VALU ------------
__global__ void proj_small(const float* __restrict__ X, const float* __restrict__ W,
                           const float* __restrict__ b, float* __restrict__ Y,
                           int Nrows, int K, int M) {
  int t = blockIdx.x * blockDim.x + threadIdx.x;
  if (t >= Nrows * M) return;
  int n = t / M, m = t % M;
  float acc = b[m];
  for (int k = 0; k < K; ++k) acc += X[(size_t)n * K + k] * W[(size_t)k * M + m];
  Y[t] = acc;
}

// ---------------- f32 WMMA GEMM: Y[N x M] = act(X[N x K] @ W[K x M] + b) ----
// Compile-time K/M: B row stride becomes an immediate offset and all address
// math strength-reduces to constant pointer increments. One wave computes a
// 32x64 slab (2 row-tiles x 4 col-tiles): each A fragment feeds 4 WMMAs, each
// B fragment feeds 2 WMMAs (8 WMMAs per K-step). Out-of-range rows/cols are
// CLAMPED for loads (unconditional loads, no EXEC juggling); stores guarded.
template <int K, int M, int ACT>
__global__ void gemm_wmma(const float* __restrict__ X, const float* __restrict__ W,
                          const float* __restrict__ bias, float* __restrict__ Y,
                          int Nrows) {
  constexpr int mT = (M + 63) / 64;      // 64-column slabs
  const int lane = threadIdx.x & 31;
  const int wave = blockIdx.x * (blockDim.x >> 5) + (threadIdx.x >> 5);
  const int nT   = (Nrows + 31) >> 5;    // 32-row slabs
  if (wave >= nT * mT) return;
  const int tm = wave % mT, tn = wave / mT;
  const int row0 = tn << 5, col0 = tm << 6;
  const int half = lane >> 4, r = lane & 15;

  int aRow0 = row0 + r;      if (aRow0 >= Nrows) aRow0 = Nrows - 1;  // clamp
  int aRow1 = row0 + 16 + r; if (aRow1 >= Nrows) aRow1 = Nrows - 1;  // clamp
  int bc[4];
#pragma unroll
  for (int j = 0; j < 4; ++j) {
    int cj = col0 + 16 * j + r;
    bc[j] = cj < M ? cj : M - 1;                                     // clamp
  }

  const float* x0 = X + (size_t)aRow0 * K;
  const float* x1 = X + (size_t)aRow1 * K;
  v8f acc[2][4] = {};
#pragma unroll 2
  for (int k = 0; k < K; k += 4) {
    v2f a0 = *(const v2f*)(x0 + k + 2 * half);    // 8B loads
    v2f a1 = *(const v2f*)(x1 + k + 2 * half);
    const float* w0 = W + (size_t)(k + 2 * half) * M;
#pragma unroll
    for (int j = 0; j < 4; ++j) {
      v2f b;
      b.x = w0[bc[j]];
      b.y = w0[bc[j] + M];                        // immediate offset (M*4 B)
      acc[0][j] = __builtin_amdgcn_wmma_f32_16x16x4_f32(
          false, a0, false, b, (short)0, acc[0][j], false, false);
      acc[1][j] = __builtin_amdgcn_wmma_f32_16x16x4_f32(
          false, a1, false, b, (short)0, acc[1][j], false, false);
    }
  }

#pragma unroll
  for (int j = 0; j < 4; ++j) {
    int col = col0 + 16 * j + r;
    if (col >= M) continue;
    float bb = bias[col];
#pragma unroll
    for (int rt = 0; rt < 2; ++rt) {
      float* yb = Y + (size_t)(row0 + 16 * rt) * M + col;
#pragma unroll
      for (int i = 0; i < 8; ++i) {
        int row = row0 + 16 * rt + i + 8 * half;
        if (row >= Nrows) continue;
        float v = acc[rt][j][i] + bb;
        if (ACT) v = leaky(v);
        yb[(size_t)(i + 8 * half) * M] = v;
      }
    }
  }
}

// ---------------- per-layer scratch init ------------------------------------
__global__ void init_buffers(float* __restrict__ agg, float* __restrict__ m,
                             float* __restrict__ den, int N) {
  int t = blockIdx.x * blockDim.x + threadIdx.x;
  if (t < N * HCQ) agg[t] = 0.f;
  if (t < N * NHD) { m[t] = -3.402823466e38f; den[t] = 0.f; }
}

// edge-embedding for one (edge, head): ec[0..15], b128 loads of We rows
__device__ __forceinline__ void edge_embed(const float* __restrict__ ea, int e,
                                           const float* __restrict__ Wc,
                                           const float* __restrict__ bc,
                                           float* __restrict__ ec) {
  float a[FED];
#pragma unroll
  for (int j = 0; j < FED; ++j) a[j] = ea[(size_t)e * FED + j];
#pragma unroll
  for (int c = 0; c < CCH; c += 4) *(float4*)&ec[c] = *(const float4*)(bc + c);
#pragma unroll
  for (int j = 0; j < FED; ++j) {
    float aj = a[j];
    const float4* wr = (const float4*)(Wc + j * HCQ);
#pragma unroll
    for (int cc = 0; cc < 4; ++cc) {
      float4 w = wr[cc];
      ec[cc * 4 + 0] += aj * w.x;
      ec[cc * 4 + 1] += aj * w.y;
      ec[cc * 4 + 2] += aj * w.z;
      ec[cc * 4 + 3] += aj * w.w;
    }
  }
}

__device__ __forceinline__ void load16(const float* __restrict__ p, float* __restrict__ d) {
#pragma unroll
  for (int c = 0; c < CCH; c += 4) *(float4*)&d[c] = *(const float4*)(p + c);
}

// ---------------- edge pass 1: segment max of attention scores --------------
__global__ void edge_pass1(const int* __restrict__ src, const int* __restrict__ dst,
                           const float* __restrict__ ea, const float* __restrict__ We,
                           const float* __restrict__ be, const float* __restrict__ q,
                           const float* __restrict__ k, float* __restrict__ m, int E) {
  int t = blockIdx.x * blockDim.x + threadIdx.x;
  if (t >= E * NHD) return;
  int e = t >> 3, h = t & 7;
  int s = src[e], d = dst[e];
  float ec[CCH], qv[CCH], kv[CCH];
  edge_embed(ea, e, We + h * CCH, be + h * CCH, ec);
  load16(q + (size_t)d * HCQ + h * CCH, qv);
  load16(k + (size_t)s * HCQ + h * CCH, kv);
  float acc = 0.f;
#pragma unroll
  for (int c = 0; c < CCH; ++c) acc += qv[c] * (kv[c] + ec[c]);
  atomicMaxF32(m + (size_t)d * NHD + h, acc * 0.25f);  // 1/sqrt(C)=0.25
}

// ---------------- edge pass 2: denom + unnormalized message scatter ---------
__global__ void edge_pass2(const int* __restrict__ src, const int* __restrict__ dst,
                           const float* __restrict__ ea, const float* __restrict__ We,
                           const float* __restrict__ be, const float* __restrict__ q,
                           const float* __restrict__ k, const float* __restrict__ v,
                           const float* __restrict__ m, float* __restrict__ den,
                           float* __restrict__ agg, int E) {
  int t = blockIdx.x * blockDim.x + threadIdx.x;
  if (t >= E * NHD) return;
  int e = t >> 3, h = t & 7;
  int s = src[e], d = dst[e];
  float ec[CCH], qv[CCH], kv[CCH], vv[CCH];
  edge_embed(ea, e, We + h * CCH, be + h * CCH, ec);
  load16(q + (size_t)d * HCQ + h * CCH, qv);
  load16(k + (size_t)s * HCQ + h * CCH, kv);
  load16(v + (size_t)s * HCQ + h * CCH, vv);
  float acc = 0.f;
#pragma unroll
  for (int c = 0; c < CCH; ++c) acc += qv[c] * (kv[c] + ec[c]);
  float ex = __expf(acc * 0.25f - m[(size_t)d * NHD + h]);
  atomicAdd(den + (size_t)d * NHD + h, ex);
  float* ap = agg + (size_t)d * HCQ + h * CCH;
#pragma unroll
  for (int c = 0; c < CCH; ++c) atomicAdd(ap + c, ex * (vv[c] + ec[c]));
}

// ---------------- node finalize: normalize + skip + LeakyReLU ---------------
__global__ void finalize_nodes(const float* __restrict__ agg, const float* __restrict__ den,
                               const float* __restrict__ skip, float* __restrict__ h, int N) {
  int t = blockIdx.x * blockDim.x + threadIdx.x;
  if (t >= N * HCQ) return;
  int n = t >> 7;
  int head = (t >> 4) & 7;
  float dd = den[(size_t)n * NHD + head];
  float v = (dd > 0.f ? agg[t] / dd : 0.f) + skip[t];
  h[t] = leaky(v);
}

// ---------------- final 500->1 dot, one wave per node -----------------------
__global__ void dot_out(const float* __restrict__ T, const float* __restrict__ Wre,
                        const float* __restrict__ bre, float* __restrict__ out,
                        int N, int K) {
  int wave = (blockIdx.x * blockDim.x + threadIdx.x) >> 5;
  int lane = threadIdx.x & 31;
  if (wave >= N) return;
  const float* tp = T + (size_t)wave * K;
  float acc = 0.f;
  for (int kk = lane; kk < K; kk += 32) acc += tp[kk] * Wre[kk];
#pragma unroll
  for (int off = 16; off; off >>= 1) acc += __shfl_xor(acc, off, 32);
  if (lane == 0) out[wave] = acc + bre[0];
}

extern "C" void kernel_launch(void* const* d_in, const int* in_sizes, int n_in,
                              void* d_out, int out_size, void* d_ws, size_t ws_size,
                              hipStream_t stream) {
  (void)n_in; (void)out_size; (void)ws_size;
  const float* x   = (const float*)d_in[0];
  const int*   ei  = (const int*)  d_in[1];
  const float* ea  = (const float*)d_in[2];
  const float* Wq1 = (const float*)d_in[3];  const float* bq1 = (const float*)d_in[4];
  const float* Wk1 = (const float*)d_in[5];  const float* bk1 = (const float*)d_in[6];
  const float* Wv1 = (const float*)d_in[7];  const float* bv1 = (const float*)d_in[8];
  const float* We1 = (const float*)d_in[9];  const float* be1 = (const float*)d_in[10];
  const float* Ws1 = (const float*)d_in[11]; const float* bs1 = (const float*)d_in[12];
  const float* Wqr = (const float*)d_in[13]; const float* bqr = (const float*)d_in[14];
  const float* Wkr = (const float*)d_in[15]; const float* bkr = (const float*)d_in[16];
  const float* Wvr = (const float*)d_in[17]; const float* bvr = (const float*)d_in[18];
  const float* Wer = (const float*)d_in[19]; const float* ber = (const float*)d_in[20];
  const float* Wsr = (const float*)d_in[21]; const float* bsr = (const float*)d_in[22];
  const float* Wr1 = (const float*)d_in[23]; const float* br1 = (const float*)d_in[24];
  const float* Wrm = (const float*)d_in[25]; const float* brm = (const float*)d_in[26];
  const float* Wre = (const float*)d_in[27]; const float* bre = (const float*)d_in[28];
  float* out = (float*)d_out;

  const int N  = in_sizes[0] / 2;   // FN = 2
  const int E  = in_sizes[1] / 2;
  const int* srcI = ei;
  const int* dstI = ei + E;

  // workspace layout (floats)
  float* ws = (float*)d_ws;
  const size_t NHCs = (size_t)N * HCQ;
  float* qb  = ws;                  // N*128
  float* kb  = qb  + NHCs;          // N*128
  float* vb  = kb  + NHCs;          // N*128
  float* sk  = vb  + NHCs;          // N*128
  float* agg = sk  + NHCs;          // N*128
  float* mb  = agg + NHCs;          // N*8
  float* den = mb  + (size_t)N * NHD;   // N*8
  float* hb  = den + (size_t)N * NHD;   // N*128  (survives into reghead)
  float* t1  = ws;                      // N*RHD, reuses dead conv buffers
  float* t2  = ws + (size_t)N * RHD;    // N*RHD, reuses dead conv buffers + hb-after-t1

  const int TB = 256;
  auto cdiv = [](long a, long b) { return (int)((a + b - 1) / b); };
  const int gNodeHC = cdiv((long)N * HCQ, TB);
  const int gEdge   = cdiv((long)E * NHD, TB);
  auto gemmGrid = [&](int M) {
    long tiles = (long)((N + 31) / 32) * ((M + 63) / 64);
    return cdiv(tiles, TB / 32);
  };

  // ---- layer 1 (K = 2 projections) ----
  proj_small<<<gNodeHC, TB, 0, stream>>>(x, Wq1, bq1, qb, N, 2, HCQ);
  proj_small<<<gNodeHC, TB, 0, stream>>>(x, Wk1, bk1, kb, N, 2, HCQ);
  proj_small<<<gNodeHC, TB, 0, stream>>>(x, Wv1, bv1, vb, N, 2, HCQ);
  proj_small<<<gNodeHC, TB, 0, stream>>>(x, Ws1, bs1, sk, N, 2, HCQ);
  init_buffers<<<gNodeHC, TB, 0, stream>>>(agg, mb, den, N);
  edge_pass1<<<gEdge, TB, 0, stream>>>(srcI, dstI, ea, We1, be1, qb, kb, mb, E);
  edge_pass2<<<gEdge, TB, 0, stream>>>(srcI, dstI, ea, We1, be1, qb, kb, vb, mb, den, agg, E);
  finalize_nodes<<<gNodeHC, TB, 0, stream>>>(agg, den, sk, hb, N);

  // ---- layers 2..4 (K = 128, WMMA projections) ----
  for (int i = 0; i < 3; ++i) {
    const float* Wq = Wqr + (size_t)i * HCQ * HCQ; const float* bq = bqr + (size_t)i * HCQ;
    const float* Wk = Wkr + (size_t)i * HCQ * HCQ; const float* bk = bkr + (size_t)i * HCQ;
    const float* Wv = Wvr + (size_t)i * HCQ * HCQ; const float* bv = bvr + (size_t)i * HCQ;
    const float* We = Wer + (size_t)i * FED * HCQ; const float* be = ber + (size_t)i * HCQ;
    const float* Wsk= Wsr + (size_t)i * HCQ * HCQ; const float* bsk= bsr + (size_t)i * HCQ;
    gemm_wmma<HCQ, HCQ, 0><<<gemmGrid(HCQ), TB, 0, stream>>>(hb, Wq,  bq,  qb, N);
    gemm_wmma<HCQ, HCQ, 0><<<gemmGrid(HCQ), TB, 0, stream>>>(hb, Wk,  bk,  kb, N);
    gemm_wmma<HCQ, HCQ, 0><<<gemmGrid(HCQ), TB, 0, stream>>>(hb, Wv,  bv,  vb, N);
    gemm_wmma<HCQ, HCQ, 0><<<gemmGrid(HCQ), TB, 0, stream>>>(hb, Wsk, bsk, sk, N);
    init_buffers<<<gNodeHC, TB, 0, stream>>>(agg, mb, den, N);
    edge_pass1<<<gEdge, TB, 0, stream>>>(srcI, dstI, ea, We, be, qb, kb, mb, E);
    edge_pass2<<<gEdge, TB, 0, stream>>>(srcI, dstI, ea, We, be, qb, kb, vb, mb, den, agg, E);
    finalize_nodes<<<gNodeHC, TB, 0, stream>>>(agg, den, sk, hb, N);
  }

  // ---- regression head ----
  gemm_wmma<HCQ, RHD, 1><<<gemmGrid(RHD), TB, 0, stream>>>(hb, Wr1, br1, t1, N);
  gemm_wmma<RHD, RHD, 1><<<gemmGrid(RHD), TB, 0, stream>>>(t1, Wrm, brm, t2, N);
  dot_out<<<cdiv(N, TB / 32), TB, 0, stream>>>(t2, Wre, bre, out, N, RHD);
}